// GATv2Convolution_20641612825476
// MI455X (gfx1250) — compile-verified
//
#include <hip/hip_runtime.h>

#define IN_CHN   128
#define HEADS    8
#define OUTC     16
#define HC       128           // HEADS*OUTC
#define ALPHA    0.3f

typedef float v2f __attribute__((ext_vector_type(2)));
typedef float v8f __attribute__((ext_vector_type(8)));

// ---------------------------------------------------------------------------
// 0) init: segmax = -inf, segsum = 0, out = 0
// ---------------------------------------------------------------------------
__global__ void __launch_bounds__(256)
gat_init(float* __restrict__ segmax, float* __restrict__ segsum,
         float* __restrict__ out, int n_nodes)
{
    int tid = blockIdx.x * blockDim.x + threadIdx.x;
    int nh  = n_nodes * HEADS;
    if (tid < nh) {
        segmax[tid] = -__builtin_inff();
        segsum[tid] = 0.0f;
    }
    if (tid < n_nodes * HC) out[tid] = 0.0f;
}

// ---------------------------------------------------------------------------
// 1) node projection GEMM on f32 WMMA: g1 = x@w1, g2 = x@w2  ([N,128]@[128,128])
//    One wave32 computes one 16x16 output tile, K swept 4 at a time with
//    V_WMMA_F32_16X16X4_F32.
//    A (16x4 f32) layout: lanes 0-15 hold M=lane, v0=K0,v1=K1; lanes 16-31: K2,K3.
//    B (4x16 f32) layout: v0 = rows K0(l0-15)|K2(l16-31), v1 = K1|K3.
//    C/D (16x16 f32): VGPR j = row M=j (lanes 0-15) and M=j+8 (lanes 16-31).
// ---------------------------------------------------------------------------
__global__ void __launch_bounds__(256)
node_gemm_wmma(const float* __restrict__ x,
               const float* __restrict__ w1,
               const float* __restrict__ w2,
               float* __restrict__ g1,
               float* __restrict__ g2,
               int n_nodes)
{
    const int lane = threadIdx.x & 31;
    const int wave = (blockIdx.x * blockDim.x + threadIdx.x) >> 5;

    const int mTiles      = n_nodes >> 4;   // 50000/16 = 3125 exactly
    const int nTiles      = HC / 16;        // 8
    const int tilesPerMat = mTiles * nTiles;
    if (wave >= 2 * tilesPerMat) return;

    const bool first = (wave < tilesPerMat);
    const float* __restrict__ w = first ? w1 : w2;
    float*       __restrict__ g = first ? g1 : g2;
    const int t    = first ? wave : (wave - tilesPerMat);
    const int mt   = t / nTiles;
    const int nt   = t - mt * nTiles;
    const int row0 = mt << 4;
    const int col0 = nt << 4;

    const int l15 = lane & 15;
    const int hi  = lane >> 4;              // 0: K pair {0,1}; 1: K pair {2,3}

    const float* __restrict__ arow = x + (size_t)(row0 + l15) * IN_CHN + 2 * hi;
    const float* __restrict__ bcol = w + (size_t)(2 * hi) * HC + col0 + l15;

    v8f c = {0.f, 0.f, 0.f, 0.f, 0.f, 0.f, 0.f, 0.f};
    #pragma unroll 4
    for (int k0 = 0; k0 < IN_CHN; k0 += 4) {
        v2f a, b;
        a.x = arow[k0];                       // A[M=l15][k0 + 2*hi]
        a.y = arow[k0 + 1];                   // A[M=l15][k0 + 2*hi + 1]
        b.x = bcol[(size_t)k0 * HC];          // B[k0 + 2*hi][col0+l15]
        b.y = bcol[(size_t)(k0 + 1) * HC];    // B[k0 + 2*hi + 1][col0+l15]
        c = __builtin_amdgcn_wmma_f32_16x16x4_f32(
                /*neg_a=*/false, a, /*neg_b=*/false, b,
                /*c_mod=*/(short)0, c, /*reuse_a=*/false, /*reuse_b=*/false);
    }

    float* __restrict__ gout = g + (size_t)(row0 + hi * 8) * HC + col0 + l15;
    #pragma unroll
    for (int j = 0; j < 8; ++j)
        gout[(size_t)j * HC] = c[j];
}

// ---------------------------------------------------------------------------
// f32 atomic max via ordered-int trick (correct for non-NaN, init = -inf)
// ---------------------------------------------------------------------------
__device__ __forceinline__ void atomicMaxF32(float* addr, float v)
{
    if (v >= 0.0f) atomicMax((int*)addr, __float_as_int(v));
    else           atomicMin((unsigned int*)addr, __float_as_uint(v));
}

// ---------------------------------------------------------------------------
// 2) per-edge logits + segment max. 8 threads per edge (thread == head);
//    each thread reduces its 16 contiguous channels (4 x float4, coalesced).
// ---------------------------------------------------------------------------
__global__ void __launch_bounds__(256)
edge_logits(const float* __restrict__ g1, const float* __restrict__ g2,
            const float* __restrict__ avec,
            const int* __restrict__ src, const int* __restrict__ tgt,
            float* __restrict__ logits, float* __restrict__ segmax,
            int n_edges)
{
    int t = blockIdx.x * blockDim.x + threadIdx.x;
    int e = t >> 3;
    int h = t & 7;
    if (e >= n_edges) return;

    int s = src[e], d = tgt[e];
    const float4* __restrict__ hi = (const float4*)(g1 + (size_t)s * HC + h * OUTC);
    const float4* __restrict__ hj = (const float4*)(g2 + (size_t)d * HC + h * OUTC);
    const float4* __restrict__ av = (const float4*)(avec + h * OUTC);

    float acc = 0.0f;
    #pragma unroll
    for (int q = 0; q < 4; ++q) {
        float4 u = hi[q], v = hj[q], w = av[q];
        float s0 = u.x + v.x, s1 = u.y + v.y, s2 = u.z + v.z, s3 = u.w + v.w;
        s0 = s0 > 0.f ? s0 : ALPHA * s0;
        s1 = s1 > 0.f ? s1 : ALPHA * s1;
        s2 = s2 > 0.f ? s2 : ALPHA * s2;
        s3 = s3 > 0.f ? s3 : ALPHA * s3;
        acc += s0 * w.x + s1 * w.y + s2 * w.z + s3 * w.w;
    }
    logits[(size_t)e * HEADS + h] = acc;
    atomicMaxF32(&segmax[(size_t)d * HEADS + h], acc);
}

// ---------------------------------------------------------------------------
// 3) exp(logit - segmax[tgt]) + segment sum
// ---------------------------------------------------------------------------
__global__ void __launch_bounds__(256)
edge_exp(const float* __restrict__ logits, const float* __restrict__ segmax,
         const int* __restrict__ tgt,
         float* __restrict__ expv, float* __restrict__ segsum, int n_edges)
{
    int t = blockIdx.x * blockDim.x + threadIdx.x;
    int e = t >> 3;
    int h = t & 7;
    if (e >= n_edges) return;

    int d   = tgt[e];
    float m = segmax[(size_t)d * HEADS + h];      // always finite for a used tgt
    float v = expf(logits[(size_t)e * HEADS + h] - m);
    expv[(size_t)e * HEADS + h] = v;
    atomicAdd(&segsum[(size_t)d * HEADS + h], v);
}

// ---------------------------------------------------------------------------
// 4) attn = exp/segsum[tgt]; out[tgt] += g1[src] * attn   (32 threads/edge,
//    4 channels each -> 4 x global f32 atomic adds)
// ---------------------------------------------------------------------------
__global__ void __launch_bounds__(256)
edge_scatter(const float* __restrict__ g1,
             const float* __restrict__ expv, const float* __restrict__ segsum,
             const int* __restrict__ src, const int* __restrict__ tgt,
             float* __restrict__ out, int n_edges)
{
    int t = blockIdx.x * blockDim.x + threadIdx.x;
    int e = t >> 5;
    int q = t & 31;              // channel quad: channels 4q..4q+3
    if (e >= n_edges) return;

    int h = q >> 2;              // head = quad/4
    int s = src[e], d = tgt[e];
    float attn = expv[(size_t)e * HEADS + h] / segsum[(size_t)d * HEADS + h];

    float4 hv = *(const float4*)(g1 + (size_t)s * HC + q * 4);
    float* o  = out + (size_t)d * HC + q * 4;
    atomicAdd(o + 0, hv.x * attn);
    atomicAdd(o + 1, hv.y * attn);
    atomicAdd(o + 2, hv.z * attn);
    atomicAdd(o + 3, hv.w * attn);
}

// ---------------------------------------------------------------------------
extern "C" void kernel_launch(void* const* d_in, const int* in_sizes, int n_in,
                              void* d_out, int out_size, void* d_ws, size_t ws_size,
                              hipStream_t stream)
{
    const float* x    = (const float*)d_in[0];
    const float* w1   = (const float*)d_in[1];
    const float* w2   = (const float*)d_in[2];
    const float* avec = (const float*)d_in[3];
    const int*   src  = (const int*)d_in[4];
    const int*   tgt  = (const int*)d_in[5];
    float*       out  = (float*)d_out;

    const int n_nodes = in_sizes[0] / IN_CHN;   // 50000
    const int n_edges = in_sizes[4];            // 1600000

    // workspace partition (floats)
    float* ws     = (float*)d_ws;
    float* g1     = ws;  ws += (size_t)n_nodes * HC;
    float* g2     = ws;  ws += (size_t)n_nodes * HC;
    float* logits = ws;  ws += (size_t)n_edges * HEADS;
    float* expv   = ws;  ws += (size_t)n_edges * HEADS;
    float* segmax = ws;  ws += (size_t)n_nodes * HEADS;
    float* segsum = ws;  ws += (size_t)n_nodes * HEADS;

    const int blk = 256;

    // 0) init accumulators / output
    {
        int total  = n_nodes * HC;
        int blocks = (total + blk - 1) / blk;
        gat_init<<<blocks, blk, 0, stream>>>(segmax, segsum, out, n_nodes);
    }
    // 1) node GEMMs on WMMA
    {
        int waves   = 2 * (n_nodes >> 4) * (HC / 16);
        int threads = waves * 32;
        int blocks  = (threads + blk - 1) / blk;
        node_gemm_wmma<<<blocks, blk, 0, stream>>>(x, w1, w2, g1, g2, n_nodes);
    }
    // 2) logits + segment max
    {
        long long threads = (long long)n_edges * HEADS;
        int blocks = (int)((threads + blk - 1) / blk);
        edge_logits<<<blocks, blk, 0, stream>>>(g1, g2, avec, src, tgt,
                                                logits, segmax, n_edges);
    }
    // 3) exp + segment sum
    {
        long long threads = (long long)n_edges * HEADS;
        int blocks = (int)((threads + blk - 1) / blk);
        edge_exp<<<blocks, blk, 0, stream>>>(logits, segmax, tgt,
                                             expv, segsum, n_edges);
    }
    // 4) weighted scatter-sum
    {
        long long threads = (long long)n_edges * 32;
        int blocks = (int)((threads + blk - 1) / blk);
        edge_scatter<<<blocks, blk, 0, stream>>>(g1, expv, segsum, src, tgt,
                                                 out, n_edges);
    }
}